// ControlNorm2DLoop_57707180589849
// MI455X (gfx1250) — compile-verified
//
#include <hip/hip_runtime.h>
#include <math.h>

// ControlNorm2DLoop for MI455X (gfx1250, wave32).
// x: [64, 256, 64, 64] f32.  Bandwidth-bound: ~768MB traffic @ 23.3TB/s ~ 33us.
// Pass 1: per-(n,c) sum / sumsq, cross-lane reduce on the WMMA pipe.
// Pass 2: sequential EMA scan over N (256 channels in parallel).
// Pass 3: streaming normalize with nontemporal float4 traffic.

#define AFWD 0.999f
#define EPS  1e-5f
#define NB   64
#define NC   256
#define HW   4096
#define NP   (NB * NC)          // 16384 (n,c) planes

typedef __attribute__((ext_vector_type(2))) float v2f;
typedef __attribute__((ext_vector_type(4))) float v4f;
typedef __attribute__((ext_vector_type(8))) float v8f;

// Full 32-lane sum using V_WMMA_F32_16X16X4_F32.
// A (16x4 f32, layout per ISA 7.12.2): lane L<16  -> A[L][0]   = val, A[L][1]   = 0
//                                      lane L>=16 -> A[L-16][2] = val, A[L-16][3] = 0
// B = ones(4x16) (all-ones is independent of B's VGPR striping convention), so
// D[M][N] = val(lane M) + val(lane M+16) for every N.
// D layout: VGPR r holds row M=r (lanes 0-15) and M=r+8 (lanes 16-31), so summing
// the 8 D VGPRs gives rows 0-7 in the low half, rows 8-15 in the high half;
// one xor-16 shuffle-add yields the full wave sum broadcast to all lanes.
static __device__ __forceinline__ float wave_sum_wmma(float val) {
  v2f a; a.x = val;  a.y = 0.0f;
  v2f b; b.x = 1.0f; b.y = 1.0f;
  v8f c = {};
  v8f d = __builtin_amdgcn_wmma_f32_16x16x4_f32(
      /*neg_a=*/false, a, /*neg_b=*/false, b,
      /*c_mod=*/(short)0, c, /*reuse_a=*/false, /*reuse_b=*/false);
  float t = ((d[0] + d[1]) + (d[2] + d[3])) + ((d[4] + d[5]) + (d[6] + d[7]));
  t += __shfl_xor(t, 16, 32);
  return t;
}

__global__ __launch_bounds__(256)
void cn2d_moments_kernel(const float* __restrict__ x,
                         float* __restrict__ mu,
                         float* __restrict__ var) {
  const int lane = threadIdx.x & 31;
  const int wave = threadIdx.x >> 5;
  const int p    = blockIdx.x * 8 + wave;              // (n,c) plane index
  const v4f* __restrict__ xp = (const v4f*)(x + (size_t)p * HW);

  float s = 0.0f, q = 0.0f;
  #pragma unroll 4
  for (int it = 0; it < 32; ++it) {                    // 32 lanes * 32 it * 4 = 4096
    v4f v = xp[it * 32 + lane];
    s += (v.x + v.y) + (v.z + v.w);
    q = fmaf(v.x, v.x, q);
    q = fmaf(v.y, v.y, q);
    q = fmaf(v.z, v.z, q);
    q = fmaf(v.w, v.w, q);
  }

  const float sum   = wave_sum_wmma(s);
  const float sumsq = wave_sum_wmma(q);

  if (lane == 0) {
    const float m = sum * (1.0f / HW);
    mu[p]  = m;
    var[p] = fmaf(-m, m, sumsq * (1.0f / HW));         // E[x^2] - mu^2
  }
}

// Sequential EMA scan over the batch dim; one thread per channel.
// Emits the *pre-update* carry (m, 1/sqrt(v+eps)) used to normalize sample n.
__global__ __launch_bounds__(256)
void cn2d_scan_kernel(const float* __restrict__ mu,
                      const float* __restrict__ var,
                      float* __restrict__ m_arr,
                      float* __restrict__ inv_arr) {
  const int c = threadIdx.x;                           // 0..255
  float m = 0.0f, v = 1.0f;
  for (int n = 0; n < NB; ++n) {
    const int idx = n * NC + c;
    m_arr[idx]   = m;
    inv_arr[idx] = 1.0f / sqrtf(v + EPS);
    const float d = mu[idx] - m;
    v = AFWD * v + (1.0f - AFWD) * var[idx] + (AFWD * (1.0f - AFWD)) * d * d;
    m = fmaf(1.0f - AFWD, d, m);
  }
}

__global__ __launch_bounds__(256)
void cn2d_normalize_kernel(const float* __restrict__ x,
                           const float* __restrict__ m_arr,
                           const float* __restrict__ inv_arr,
                           float* __restrict__ out) {
  const int b = blockIdx.x;                            // (n,c) plane index
  const float m   = m_arr[b];                          // block-uniform -> s_load
  const float inv = inv_arr[b];
  const v4f* __restrict__ xp = (const v4f*)(x   + (size_t)b * HW);
  v4f*       __restrict__ op = (v4f*)      (out + (size_t)b * HW);
  #pragma unroll
  for (int i = 0; i < 4; ++i) {                        // 256 thr * 4 it * 4 = 4096
    const int idx = i * 256 + threadIdx.x;
    v4f v = __builtin_nontemporal_load(xp + idx);      // never re-read
    v4f r;
    r.x = (v.x - m) * inv;
    r.y = (v.y - m) * inv;
    r.z = (v.z - m) * inv;
    r.w = (v.w - m) * inv;
    __builtin_nontemporal_store(r, op + idx);          // never re-read; keep L2 clean
  }
}

extern "C" void kernel_launch(void* const* d_in, const int* in_sizes, int n_in,
                              void* d_out, int out_size, void* d_ws, size_t ws_size,
                              hipStream_t stream) {
  const float* x   = (const float*)d_in[0];
  float*       out = (float*)d_out;
  float*       ws  = (float*)d_ws;                     // needs 4*NP floats = 256KB
  float* mu      = ws;
  float* var     = ws + NP;
  float* m_arr   = ws + 2 * NP;
  float* inv_arr = ws + 3 * NP;

  cn2d_moments_kernel  <<<NP / 8, 256, 0, stream>>>(x, mu, var);
  cn2d_scan_kernel     <<<1,      NC,  0, stream>>>(mu, var, m_arr, inv_arr);
  cn2d_normalize_kernel<<<NP,     256, 0, stream>>>(x, m_arr, inv_arr, out);
}